// FuzzyLabelLoss_90598040142226
// MI455X (gfx1250) — compile-verified
//
#include <hip/hip_runtime.h>
#include <math.h>

#define NUM_CLASSES 35
#define BLOCK 256
#define GRID 2048

typedef __attribute__((ext_vector_type(2))) float v2f;
typedef __attribute__((ext_vector_type(8))) float v8f;

// Reduce 256 floats in LDS using V_WMMA_F32_16X16X4_F32.
// A = ones(16x4) so D[m,n] = sum_k B[k,n]; summing row 0 across its 16
// columns yields the exact sum of all 64 B elements per WMMA, independent
// of the B-operand lane/VGPR packing. Accumulate C across 4 chunks.
// Must be called with the full wave active (EXEC all 1s). Lane 0 returns sum.
__device__ __forceinline__ float wmma_reduce256(const float* lds, int lane) {
    v2f a; a.x = 1.0f; a.y = 1.0f;           // ones 16x4 A-matrix (2 VGPRs)
    v8f c = {};
#pragma unroll
    for (int i = 0; i < 4; ++i) {
        v2f b;
        b.x = lds[i * 64 + lane];            // 64 values per WMMA, packed
        b.y = lds[i * 64 + 32 + lane];       // arbitrarily into the 2 B VGPRs
        c = __builtin_amdgcn_wmma_f32_16x16x4_f32(
                /*neg_a=*/false, a, /*neg_b=*/false, b,
                /*c_mod=*/(short)0, c, /*reuse_a=*/false, /*reuse_b=*/false);
    }
    // c[0] lanes 0..15 hold D[0][n] (column sums). Sum within the 16-group.
    float v = c[0];
    v += __shfl_xor(v, 8, 32);
    v += __shfl_xor(v, 4, 32);
    v += __shfl_xor(v, 2, 32);
    v += __shfl_xor(v, 1, 32);
    return v;                                 // lane 0: total of 256 inputs
}

__global__ void __launch_bounds__(BLOCK)
fuzzy_ce_pass1(const float* __restrict__ logits,
               const int* __restrict__ targets,
               const int* __restrict__ turns,
               float* __restrict__ partials,
               int B) {
    __shared__ float lds[BLOCK];

    float acc = 0.0f;
    const int stride = gridDim.x * blockDim.x;
    for (int row = blockIdx.x * blockDim.x + threadIdx.x; row < B; row += stride) {
        const float* __restrict__ x = logits + (long)row * NUM_CLASSES;
        const int t = targets[row];

        float xs[NUM_CLASSES];
        float m  = -INFINITY;
        float xt = 0.0f;                      // logit of the target class
#pragma unroll
        for (int k = 0; k < NUM_CLASSES; ++k) {
            const float v = x[k];
            xs[k] = v;
            m = fmaxf(m, v);
            xt = (k == t) ? v : xt;           // cndmask; avoids dynamic index
        }
        float s = 0.0f;
#pragma unroll
        for (int k = 0; k < NUM_CLASSES; ++k) {
            s += __expf(xs[k] - m);
        }
        const float loss = (m + __logf(s)) - xt;   // -log_softmax[target]

        // weight: clamp(0.7 + 0.3*(turn-6)/6, 0.7, 1.0)
        const float tf = (float)turns[row];
        const float w  = fminf(1.0f, fmaxf(0.7f, 0.7f + 0.05f * (tf - 6.0f)));

        acc += loss * w;
    }

    lds[threadIdx.x] = acc;
    __syncthreads();

    if (threadIdx.x < 32) {                   // wave 0, EXEC all 1s
        const float v = wmma_reduce256(lds, (int)threadIdx.x);
        if (threadIdx.x == 0) partials[blockIdx.x] = v;
    }
}

__global__ void __launch_bounds__(BLOCK)
fuzzy_ce_pass2(const float* __restrict__ partials,
               int n,
               float* __restrict__ out,
               float invB) {
    __shared__ float lds[BLOCK];

    float acc = 0.0f;
    for (int i = threadIdx.x; i < n; i += BLOCK) acc += partials[i];
    lds[threadIdx.x] = acc;
    __syncthreads();

    if (threadIdx.x < 32) {
        const float v = wmma_reduce256(lds, (int)threadIdx.x);
        if (threadIdx.x == 0) out[0] = v * invB;
    }
}

extern "C" void kernel_launch(void* const* d_in, const int* in_sizes, int n_in,
                              void* d_out, int out_size, void* d_ws, size_t ws_size,
                              hipStream_t stream) {
    const float* logits  = (const float*)d_in[0];
    const int*   targets = (const int*)d_in[1];
    const int*   turns   = (const int*)d_in[2];
    float*       outp    = (float*)d_out;
    float*       partials = (float*)d_ws;     // GRID floats (8 KB)

    const int B = in_sizes[1];                // number of rows

    fuzzy_ce_pass1<<<GRID, BLOCK, 0, stream>>>(logits, targets, turns, partials, B);
    fuzzy_ce_pass2<<<1, BLOCK, 0, stream>>>(partials, GRID, outp, 1.0f / (float)B);
}